// AdaptiveMaskFusion_11957188952232
// MI455X (gfx1250) — compile-verified
//
#include <hip/hip_runtime.h>
#include <math.h>

#define ALPHA   0.42f
#define HGT     1024
#define WID     1024
#define NB      16
#define NS      3
#define NPLANE  48
#define TILE    32
#define HALO    3
#define PADW    (TILE + 2*HALO)    /* 38 */
#define PADDED  (PADW*PADW)        /* 1444 */
#define TPB     256
#define TILES_X (WID/TILE)         /* 32 */
#define TILES_Y (HGT/TILE)         /* 32 */
#define TILES_PER_PLANE (TILES_X*TILES_Y) /* 1024 */

typedef float vfloat4 __attribute__((ext_vector_type(4)));

/* ---- CDNA5 async global->LDS path (guarded; bridge doc warns of toolchain skew) ---- */
#if __has_builtin(__builtin_amdgcn_global_load_async_to_lds_b32)
#define USE_ASYNC_LDS 1
#define ASYNC_LOAD_B32(gptr, lptr)                                            \
  __builtin_amdgcn_global_load_async_to_lds_b32(                              \
      (__attribute__((address_space(1))) int*)(gptr),                         \
      (__attribute__((address_space(3))) int*)(lptr), 0, 0)
#if __has_builtin(__builtin_amdgcn_s_wait_asynccnt)
#define WAIT_ASYNC() __builtin_amdgcn_s_wait_asynccnt(0)
#else
#define WAIT_ASYNC() asm volatile("s_wait_asynccnt 0" ::: "memory")
#endif
#else
#define USE_ASYNC_LDS 0
#endif

__device__ __forceinline__ float fast_sigmoid(float x) {
  return 1.0f / (1.0f + __expf(-x));
}

/* ---------------- Kernel 1: per-tile score partials ---------------- */
__global__ __launch_bounds__(TPB) void k_score(const float* __restrict__ logits,
                                               float* __restrict__ partials) {
  __shared__ float sm[PADDED];       // sigmoid tile + halo (zero padded outside image)
  __shared__ float rs[PADW * TILE];  // 7-wide row sums; reused for block reduction

  const int tx = blockIdx.x, ty = blockIdx.y, plane = blockIdx.z;
  const int tid = threadIdx.x;
  const float* src = logits + ((size_t)plane << 20);
  const int gx0 = tx * TILE - HALO;
  const int gy0 = ty * TILE - HALO;

#if USE_ASYNC_LDS
  for (int idx = tid; idx < PADDED; idx += TPB) {
    const int py = idx / PADW, px = idx - py * PADW;
    const int gy = gy0 + py, gx = gx0 + px;
    if ((unsigned)gy < HGT && (unsigned)gx < WID) {
      ASYNC_LOAD_B32(src + (size_t)gy * WID + gx, &sm[idx]);
    }
  }
  WAIT_ASYNC();
  __syncthreads();
  for (int idx = tid; idx < PADDED; idx += TPB) {
    const int py = idx / PADW, px = idx - py * PADW;
    const int gy = gy0 + py, gx = gx0 + px;
    float v = 0.0f;
    if ((unsigned)gy < HGT && (unsigned)gx < WID) v = fast_sigmoid(sm[idx]);
    sm[idx] = v;
  }
#else
  for (int idx = tid; idx < PADDED; idx += TPB) {
    const int py = idx / PADW, px = idx - py * PADW;
    const int gy = gy0 + py, gx = gx0 + px;
    float v = 0.0f;
    if ((unsigned)gy < HGT && (unsigned)gx < WID)
      v = fast_sigmoid(src[(size_t)gy * WID + gx]);
    sm[idx] = v;
  }
#endif
  __syncthreads();

  /* separable 7-wide horizontal box sums: rs[py][lx], py in [0,38), lx in [0,32) */
  for (int idx = tid; idx < PADW * TILE; idx += TPB) {
    const int py = idx / TILE, lx = idx - py * TILE;
    const float* r = &sm[py * PADW + lx];
    rs[idx] = r[0] + r[1] + r[2] + r[3] + r[4] + r[5] + r[6];
  }
  __syncthreads();

  float acc = 0.0f;
#pragma unroll
  for (int k = 0; k < (TILE * TILE) / TPB; ++k) {
    const int t = tid + k * TPB;
    const int ly = t >> 5, lx = t & 31;
    const float* c = &sm[(ly + HALO) * PADW + (lx + HALO)];
    const float a00 = c[-PADW - 1], a01 = c[-PADW], a02 = c[-PADW + 1];
    const float a10 = c[-1],        a11 = c[0],     a12 = c[1];
    const float a20 = c[PADW - 1],  a21 = c[PADW],  a22 = c[PADW + 1];
    const float gxs = (a00 - a02) + 2.0f * (a10 - a12) + (a20 - a22);
    const float gys = (a00 + 2.0f * a01 + a02) - (a20 + 2.0f * a21 + a22);
    const float lp  = a01 + a10 + a12 + a21 - 4.0f * a11;
    const float edge = sqrtf(gxs * gxs + gys * gys) + 0.5f * fabsf(lp);

    float s7 = 0.0f;
#pragma unroll
    for (int dy = 0; dy < 7; ++dy) s7 += rs[(ly + dy) * TILE + lx];
    const float mean = s7 * (1.0f / 49.0f);
    const float diff = a11 - mean;

    /* sum_p avgpool(d)[p] == sum_q d[q]*c(q)/49 with c(q)= #windows covering q */
    const int gy = ty * TILE + ly, gx = tx * TILE + lx;
    const int cy = min(HGT - 1, gy + 3) - max(0, gy - 3) + 1;
    const int cx = min(WID - 1, gx + 3) - max(0, gx - 3) + 1;
    const float var = diff * diff * (float)(cy * cx) * (1.0f / 49.0f);

    acc += ALPHA * edge + (1.0f - ALPHA) * var;
  }
  __syncthreads();
  rs[tid] = acc;
  __syncthreads();
#pragma unroll
  for (int off = TPB >> 1; off > 0; off >>= 1) {
    if (tid < off) rs[tid] += rs[tid + off];
    __syncthreads();
  }
  if (tid == 0)
    partials[plane * TILES_PER_PLANE + (ty * TILES_X + tx)] = rs[0];
}

/* ---------------- Kernel 2: deterministic per-plane reduction ---------------- */
__global__ __launch_bounds__(256) void k_reduce(const float* __restrict__ partials,
                                                float* __restrict__ scores) {
  __shared__ float red[256];
  const int plane = blockIdx.x, tid = threadIdx.x;
  const float* p = partials + plane * TILES_PER_PLANE;
  red[tid] = p[tid] + p[tid + 256] + p[tid + 512] + p[tid + 768];
  __syncthreads();
#pragma unroll
  for (int off = 128; off > 0; off >>= 1) {
    if (tid < off) red[tid] += red[tid + off];
    __syncthreads();
  }
  if (tid == 0) scores[plane] = red[0];
}

/* ---------------- Kernel 3: 48-value weight computation ---------------- */
__global__ void k_weights(const float* __restrict__ scores,
                          const float* __restrict__ lw,
                          float* __restrict__ weights) {
  const int tid = threadIdx.x;
  if (tid >= NPLANE) return;
  const int b = tid / NS, s = tid - b * NS;
  const float l0 = lw[0], l1 = lw[1], l2 = lw[2];
  const float mx = fmaxf(l0, fmaxf(l1, l2));
  const float e0 = __expf(l0 - mx), e1 = __expf(l1 - mx), e2 = __expf(l2 - mx);
  const float esum = e0 + e1 + e2;
  const float wl = (s == 0 ? e0 : (s == 1 ? e1 : e2)) / esum;
  const float denom = scores[b * NS] + scores[b * NS + 1] + scores[b * NS + 2] + 1e-6f;
  weights[tid] = 0.5f * (scores[tid] / denom + wl);
}

/* ---------------- Kernel 4: streaming fuse (NT loads/stores + prefetch) ---------------- */
__global__ __launch_bounds__(256) void k_fuse(const float* __restrict__ logits,
                                              const float* __restrict__ weights,
                                              float* __restrict__ out) {
  const size_t PLANE4 = (size_t)(HGT * WID) / 4;   /* 262144 = 2^18 */
  const size_t total4 = (size_t)NB * PLANE4;       /* 2^22 */
  const size_t stride = (size_t)gridDim.x * blockDim.x;
  const vfloat4* in4 = (const vfloat4*)logits;
  vfloat4* out4 = (vfloat4*)out;

  for (size_t i4 = (size_t)blockIdx.x * blockDim.x + threadIdx.x; i4 < total4; i4 += stride) {
    const int b = (int)(i4 >> 18);
    const size_t r = i4 & (PLANE4 - 1);
    const vfloat4* p = in4 + ((size_t)(b * 3) << 18) + r;
    __builtin_prefetch(p + stride, 0, 1);  /* global_prefetch for the next grid-stride step */
    const float w0 = weights[b * 3 + 0];
    const float w1 = weights[b * 3 + 1];
    const float w2 = weights[b * 3 + 2];
    const vfloat4 a = __builtin_nontemporal_load(p);
    const vfloat4 c = __builtin_nontemporal_load(p + PLANE4);
    const vfloat4 d = __builtin_nontemporal_load(p + 2 * PLANE4);
    vfloat4 o;
    o.x = w0 * a.x + w1 * c.x + w2 * d.x;
    o.y = w0 * a.y + w1 * c.y + w2 * d.y;
    o.z = w0 * a.z + w1 * c.z + w2 * d.z;
    o.w = w0 * a.w + w1 * c.w + w2 * d.w;
    __builtin_nontemporal_store(o, &out4[i4]);
  }
}

extern "C" void kernel_launch(void* const* d_in, const int* in_sizes, int n_in,
                              void* d_out, int out_size, void* d_ws, size_t ws_size,
                              hipStream_t stream) {
  const float* logits = (const float*)d_in[0];
  const float* lw     = (const float*)d_in[1];
  float* ws = (float*)d_ws;
  float* partials = ws;                          /* 48*1024 floats */
  float* scores   = ws + NPLANE * TILES_PER_PLANE;
  float* weights  = scores + 64;

  dim3 g1(TILES_X, TILES_Y, NPLANE);
  k_score<<<g1, TPB, 0, stream>>>(logits, partials);
  k_reduce<<<NPLANE, 256, 0, stream>>>(partials, scores);
  k_weights<<<1, 64, 0, stream>>>(scores, lw, weights);
  k_fuse<<<8192, 256, 0, stream>>>(logits, weights, (float*)d_out);
}